// ScalableSNN_71751723647586
// MI455X (gfx1250) — compile-verified
//
#include <hip/hip_runtime.h>

// ---------------------------------------------------------------------------
// ScalableSNN forward pass for gfx1250 (MI455X).
// Memory-bound edge-list SpMM (~330 MB streamed, ~15-20 us floor @ 23.3 TB/s):
//  - edges streamed via async global->LDS DMA, double-buffered on ASYNCcnt,
//    marked non-temporal (single-use) so L2 keeps the hot atomic working set
//  - scatter-adds resolved with HW global_atomic_add_f32 in the 192 MB L2
//  - tiny motor accumulator privatized per-block in LDS (ds_add_f32)
//  - 95% of the 16M HH edges skipped via spike-sparsity branch
// WMMA is inapplicable: unstructured sparsity, no dense 16x16 tiles.
// ---------------------------------------------------------------------------

#define N_SENS 10000
#define N_HID  400000
#define N_MOT  1000
#define BETA   0.9f
#define THR    1.0f

// ---- CDNA5 async-to-LDS feature detection ---------------------------------
#if defined(__has_builtin)
#  if __has_builtin(__builtin_amdgcn_global_load_async_to_lds_b128)
#    define HAVE_ASYNC_LDS 1
#  else
#    define HAVE_ASYNC_LDS 0
#  endif
#  if __has_builtin(__builtin_amdgcn_s_wait_asynccnt)
#    define HAVE_WAIT_ASYNCCNT 1
#  else
#    define HAVE_WAIT_ASYNCCNT 0
#  endif
#else
#  define HAVE_ASYNC_LDS 0
#  define HAVE_WAIT_ASYNCCNT 0
#endif

#if HAVE_WAIT_ASYNCCNT
#  define WAIT_ASYNC(n) __builtin_amdgcn_s_wait_asynccnt(n)
#else
#  define WAIT_ASYNC(n) asm volatile("s_wait_asynccnt %0" ::"i"(n) : "memory")
#endif

// CPol temporal hint: TH[2:0]=1 -> NT (non-temporal) for single-use streams.
#define CPOL_NT 1

// Builtin signature (probe-confirmed): (addrspace(1) v4i*, addrspace(3) v4i*,
// imm offset, imm cpol).
typedef int v4i __attribute__((ext_vector_type(4)));
typedef __attribute__((address_space(1))) v4i gv4i;  // global int4
typedef __attribute__((address_space(3))) v4i lv4i;  // LDS int4

// Hardware global_atomic_add_f32 (no CAS loop).
__device__ __forceinline__ void gatomic_add(float* p, float v) {
#if defined(__HIP__)
  unsafeAtomicAdd(p, v);
#else
  atomicAdd(p, v);
#endif
}

__device__ __forceinline__ void edge_accum(int p, int q, float wt, float scale,
                                           const float* __restrict__ act,
                                           float* __restrict__ accum) {
  const float a = act[p];              // spikes are 0/1: skip dead edges
  if (a != 0.0f) gatomic_add(&accum[q], scale * wt * a);
}

// ---- Kernel 1: zero accumulators + sensory LIF ----------------------------
__global__ void __launch_bounds__(256)
snn_init(const float* __restrict__ sens_in, const float* __restrict__ sens_mem,
         float* __restrict__ sens_spk, float* __restrict__ hid_in,
         float* __restrict__ mot_in) {
  const int i = blockIdx.x * 256 + threadIdx.x;
  if (i < N_HID) hid_in[i] = 0.0f;
  if (i < N_SENS) {
    const float mem = BETA * sens_mem[i] + 5.0f * sens_in[i];
    sens_spk[i] = (mem - THR > 0.0f) ? 1.0f : 0.0f;
  }
  if (i < N_MOT) mot_in[i] = 0.0f;
}

// ---- Kernel 2/3: edge SpMM, async-staged through LDS ----------------------
// Per wave: 128-edge tiles, double-buffered. Each lane async-loads its own
// 16B chunk of pre/post/w into LDS; the DMA for tile t+1 overlaps the
// gather+atomic work for tile t (ASYNCcnt <= 3 gate, in-order retirement).
__global__ void __launch_bounds__(256)
snn_spmm(const float* __restrict__ w, const int* __restrict__ pre,
         const int* __restrict__ post, const float* __restrict__ act,
         float scale, float* __restrict__ accum, int E) {
#if HAVE_ASYNC_LDS
  __shared__ __align__(16) int   s_pre [2][8][128];
  __shared__ __align__(16) int   s_post[2][8][128];
  __shared__ __align__(16) float s_w   [2][8][128];

  const int lane = threadIdx.x & 31;      // wave32
  const int wv   = threadIdx.x >> 5;      // 8 waves / block
  const int gw   = blockIdx.x * 8 + wv;
  const int nw   = gridDim.x * 8;
  const int ntiles = E >> 7;              // 128 edges per tile

  auto issue = [&](int tile, int b) {
    const int base = (tile << 7) + (lane << 2);
    __builtin_amdgcn_global_load_async_to_lds_b128(
        (gv4i*)(pre + base),  (lv4i*)&s_pre [b][wv][lane << 2], 0, CPOL_NT);
    __builtin_amdgcn_global_load_async_to_lds_b128(
        (gv4i*)(post + base), (lv4i*)&s_post[b][wv][lane << 2], 0, CPOL_NT);
    __builtin_amdgcn_global_load_async_to_lds_b128(
        (gv4i*)(w + base),    (lv4i*)&s_w   [b][wv][lane << 2], 0, CPOL_NT);
  };

  int buf = 0;
  if (gw < ntiles) issue(gw, 0);
  for (int t = gw; t < ntiles; t += nw) {
    const int tn = t + nw;
    if (tn < ntiles) {
      issue(tn, buf ^ 1);   // prefetch next tile into the other buffer
      WAIT_ASYNC(3);        // current tile's 3 DMAs retired (in-order)
    } else {
      WAIT_ASYNC(0);
    }
    const int4   p4 = *reinterpret_cast<const int4*>  (&s_pre [buf][wv][lane << 2]);
    const int4   q4 = *reinterpret_cast<const int4*>  (&s_post[buf][wv][lane << 2]);
    const float4 w4 = *reinterpret_cast<const float4*>(&s_w   [buf][wv][lane << 2]);
    edge_accum(p4.x, q4.x, w4.x, scale, act, accum);
    edge_accum(p4.y, q4.y, w4.y, scale, act, accum);
    edge_accum(p4.z, q4.z, w4.z, scale, act, accum);
    edge_accum(p4.w, q4.w, w4.w, scale, act, accum);
    buf ^= 1;
  }
  // Remainder edges (E not a multiple of 128)
  for (int e = (ntiles << 7) + blockIdx.x * blockDim.x + threadIdx.x; e < E;
       e += gridDim.x * blockDim.x)
    edge_accum(__builtin_nontemporal_load(pre + e),
               __builtin_nontemporal_load(post + e),
               __builtin_nontemporal_load(w + e), scale, act, accum);
#else
  for (int e = blockIdx.x * blockDim.x + threadIdx.x; e < E;
       e += gridDim.x * blockDim.x)
    edge_accum(__builtin_nontemporal_load(pre + e),
               __builtin_nontemporal_load(post + e),
               __builtin_nontemporal_load(w + e), scale, act, accum);
#endif
}

// ---- Kernel 4: hidden LIF -------------------------------------------------
__global__ void __launch_bounds__(256)
snn_hid_lif(const float* __restrict__ hid_mem, const float* __restrict__ hid_in,
            float* __restrict__ hid_spk) {
  const int i = blockIdx.x * 256 + threadIdx.x;
  if (i < N_HID) {
    const float mem = BETA * hid_mem[i] + 5.0f * hid_in[i];
    hid_spk[i] = (mem - THR > 0.0f) ? 1.0f : 0.0f;
  }
}

// ---- Kernel 5: hidden->motor SpMM, LDS-privatized accumulator -------------
__global__ void __launch_bounds__(256)
snn_spmm_motor(const float* __restrict__ w, const int* __restrict__ pre,
               const int* __restrict__ post, const float* __restrict__ act,
               float* __restrict__ mot_in, int E) {
  __shared__ float acc[N_MOT];
  for (int i = threadIdx.x; i < N_MOT; i += blockDim.x) acc[i] = 0.0f;
  __syncthreads();
  for (int e = blockIdx.x * blockDim.x + threadIdx.x; e < E;
       e += gridDim.x * blockDim.x) {
    const float a = act[__builtin_nontemporal_load(pre + e)];
    if (a != 0.0f)
      atomicAdd(&acc[__builtin_nontemporal_load(post + e)],
                __builtin_nontemporal_load(w + e) * a);     // ds_add_f32
  }
  __syncthreads();
  for (int i = threadIdx.x; i < N_MOT; i += blockDim.x) {
    const float v = acc[i];
    if (v != 0.0f) gatomic_add(&mot_in[i], v);
  }
}

// ---- Kernel 6: motor LIF --------------------------------------------------
__global__ void __launch_bounds__(256)
snn_motor_lif(const float* __restrict__ mot_mem, const float* __restrict__ mot_in,
              float* __restrict__ out) {
  const int i = blockIdx.x * 256 + threadIdx.x;
  if (i < N_MOT) {
    const float mem = BETA * mot_mem[i] + 20.0f * mot_in[i];
    out[i] = (mem - THR > 0.0f) ? 1.0f : 0.0f;
  }
}

// ---------------------------------------------------------------------------
extern "C" void kernel_launch(void* const* d_in, const int* in_sizes, int n_in,
                              void* d_out, int out_size, void* d_ws, size_t ws_size,
                              hipStream_t stream) {
  const float* sensory_input = (const float*)d_in[0];
  const float* sensory_mem   = (const float*)d_in[1];
  const float* hidden_mem    = (const float*)d_in[2];
  const float* motor_mem     = (const float*)d_in[3];
  const float* hidden_prev   = (const float*)d_in[4];
  const float* w_sh          = (const float*)d_in[5];
  const float* w_hh          = (const float*)d_in[6];
  const float* w_hm          = (const float*)d_in[7];
  const int*   sh_pre        = (const int*)d_in[8];
  const int*   sh_post       = (const int*)d_in[9];
  const int*   hh_pre        = (const int*)d_in[10];
  const int*   hh_post       = (const int*)d_in[11];
  const int*   hm_pre        = (const int*)d_in[12];
  const int*   hm_post       = (const int*)d_in[13];

  const int E_SH = in_sizes[5];
  const int E_HH = in_sizes[6];
  const int E_HM = in_sizes[7];

  float* f = (float*)d_ws;                 // workspace layout (64B aligned)
  float* sens_spk = f;                     // 10000
  float* hid_in   = f + 10240;             // 400000
  float* hid_spk  = f + 410240;            // 400000
  float* mot_in   = f + 810240;            // 1000
  float* out      = (float*)d_out;         // 1000

  const int blkN = 256;
  const int gHid = (N_HID + blkN - 1) / blkN;   // 1563

  snn_init<<<gHid, blkN, 0, stream>>>(sensory_input, sensory_mem,
                                      sens_spk, hid_in, mot_in);
  snn_spmm<<<2048, blkN, 0, stream>>>(w_sh, sh_pre, sh_post, sens_spk,
                                      1.0f, hid_in, E_SH);
  snn_spmm<<<4096, blkN, 0, stream>>>(w_hh, hh_pre, hh_post, hidden_prev,
                                      0.5f, hid_in, E_HH);
  snn_hid_lif<<<gHid, blkN, 0, stream>>>(hidden_mem, hid_in, hid_spk);
  snn_spmm_motor<<<256, blkN, 0, stream>>>(w_hm, hm_pre, hm_post, hid_spk,
                                           mot_in, E_HM);
  snn_motor_lif<<<(N_MOT + blkN - 1) / blkN, blkN, 0, stream>>>(motor_mem,
                                                                mot_in, out);
}